// MoESwiGLUFFN_58798102282426
// MI455X (gfx1250) — compile-verified
//
#include <hip/hip_runtime.h>
#include <hip/hip_bf16.h>

// ---------------------------------------------------------------------------
// MoE SwiGLU FFN for gfx1250 (MI455X), fp32, V_WMMA_F32_16X16X4_F32 path.
// M=32 tiles, branch-free WMMA inner loops, async direct-to-LDS staging.
// ---------------------------------------------------------------------------

typedef __attribute__((ext_vector_type(2))) float v2f;
typedef __attribute__((ext_vector_type(8))) float v8f;

#define N_TOK 4096
#define EDIM  1024
#define HDIM  2730
#define HPAD  2816          // 22 * 128 = 11 * 256
#define NEXP  8
#define MTILE 32
#define MAX_TILES 264       // sum ceil(cnt_e/32) <= 8192/32 + 8
#define KC    256           // K-chunk staged in LDS
#define LDSS  260           // 256 + 4 pad -> conflict-free WMMA A reads

// workspace layout (bytes)
#define OFF_MOE  ((size_t)0)                                   // 4096*1024*4 = 16 MiB
#define OFF_CNT  ((size_t)N_TOK * EDIM * 4)                    // 8 ints
#define OFF_TOK  (OFF_CNT + 256)                               // 8*4096 ints
#define OFF_W    (OFF_TOK + (size_t)NEXP * N_TOK * 4)          // 8*4096 floats
#define OFF_TILE (OFF_W   + (size_t)NEXP * N_TOK * 4)          // MAX_TILES * int4
#define OFF_NT   (OFF_TILE + (size_t)MAX_TILES * 16)           // 1 int
#define OFF_H    (OFF_NT + 128)                                // 8448 * 2816 floats

__device__ __forceinline__ v8f wmma_f32_4(v2f a, v2f b, v8f c) {
    // D = A(16x4) * B(4x16) + C, fp32, wave32
    return __builtin_amdgcn_wmma_f32_16x16x4_f32(
        /*neg_a=*/false, a, /*neg_b=*/false, b,
        /*c_mod=*/(short)0, c, /*reuse_a=*/false, /*reuse_b=*/false);
}

// --- async global -> LDS staging (CDNA5 GLOBAL_LOAD_ASYNC_TO_LDS_B128) -----
#if defined(__AMDGCN__) && \
    __has_builtin(__builtin_amdgcn_global_load_async_to_lds_b128) && \
    __has_builtin(__builtin_amdgcn_s_wait_asynccnt)
#define HAVE_ASYNC_LDS 1
typedef int v4i_vs __attribute__((vector_size(16)));
__device__ __forceinline__ void stage16(float* lds_dst, const float* gsrc) {
    __builtin_amdgcn_global_load_async_to_lds_b128(
        (v4i_vs*)gsrc, (v4i_vs*)lds_dst, 0, 0);
}
__device__ __forceinline__ void stage_wait() {
    __builtin_amdgcn_s_wait_asynccnt(0);
}
#else
__device__ __forceinline__ void stage16(float* lds_dst, const float* gsrc) {
    *(float4*)lds_dst = *(const float4*)gsrc;
}
__device__ __forceinline__ void stage_wait() {}
#endif

// ---------------------------------------------------------------------------
// Kernel 1: router. One wave (32 lanes) per token: 8 logits, top-2, softmax,
// atomic compaction into per-expert token/weight lists.
// ---------------------------------------------------------------------------
__global__ __launch_bounds__(256) void moe_router(
    const float* __restrict__ x, const float* __restrict__ Wr,
    int* __restrict__ cnt, int* __restrict__ tokList, float* __restrict__ wList)
{
    int wave = threadIdx.x >> 5;
    int lane = threadIdx.x & 31;
    int n = blockIdx.x * 8 + wave;               // grid.x = 512 -> 4096 tokens
    const float* row = x + (size_t)n * EDIM;

    float acc[NEXP];
#pragma unroll
    for (int c = 0; c < NEXP; ++c) acc[c] = 0.f;
    for (int k = lane; k < EDIM; k += 32) {
        float xv = row[k];
#pragma unroll
        for (int c = 0; c < NEXP; ++c) acc[c] += xv * Wr[c * EDIM + k];
    }
#pragma unroll
    for (int c = 0; c < NEXP; ++c) {
#pragma unroll
        for (int m = 16; m >= 1; m >>= 1) acc[c] += __shfl_xor(acc[c], m, 32);
    }
    if (lane == 0) {
        int i0 = 0; float v0 = acc[0];
#pragma unroll
        for (int c = 1; c < NEXP; ++c) if (acc[c] > v0) { v0 = acc[c]; i0 = c; }
        int i1 = -1; float v1 = -3.402823466e38f;
#pragma unroll
        for (int c = 0; c < NEXP; ++c) if (c != i0 && acc[c] > v1) { v1 = acc[c]; i1 = c; }
        float m = fmaxf(v0, v1);
        float e0 = expf(v0 - m), e1 = expf(v1 - m);
        float inv = 1.f / (e0 + e1);
        int p0 = atomicAdd(&cnt[i0], 1);
        tokList[i0 * N_TOK + p0] = n; wList[i0 * N_TOK + p0] = e0 * inv;
        int p1 = atomicAdd(&cnt[i1], 1);
        tokList[i1 * N_TOK + p1] = n; wList[i1 * N_TOK + p1] = e1 * inv;
    }
}

// ---------------------------------------------------------------------------
// Kernel 2: build tile descriptors (expert, slotBase, hRowBase), M=32 tiles.
// ---------------------------------------------------------------------------
__global__ void moe_setup_tiles(const int* __restrict__ cnt,
                                int4* __restrict__ tiles, int* __restrict__ totalTiles)
{
    if (threadIdx.x == 0 && blockIdx.x == 0) {
        int t = 0, rowBase = 0;
        for (int e = 0; e < NEXP; ++e) {
            int c = cnt[e];
            int nt = (c + MTILE - 1) >> 5;
            for (int i = 0; i < nt; ++i) {
                tiles[t] = make_int4(e, i * MTILE, rowBase, 0);
                rowBase += MTILE;
                ++t;
            }
        }
        *totalTiles = t;
    }
}

// ---------------------------------------------------------------------------
// Kernel 3: pass 1. One workgroup = one 32-token tile x 128 H columns.
// 8 waves; each wave owns one 16-wide column slice, two fp32 WMMA chains
// (Wv, Wg) x two 16-row A fragments -> 4 wmma per K-step. Loader: each thread
// owns one row (8 thr/row x 8 float4), token/row pointers hoisted out of the
// K loop, async direct-to-LDS staging.
// ---------------------------------------------------------------------------
__global__ __launch_bounds__(256) void moe_pass1(
    const float* __restrict__ x,
    const float* __restrict__ Wv, const float* __restrict__ Wg,
    const float* __restrict__ bv, const float* __restrict__ bg,
    const int* __restrict__ cnt, const int* __restrict__ tokList,
    const int4* __restrict__ tiles, const int* __restrict__ totalTiles,
    float* __restrict__ hbuf)
{
    if ((int)blockIdx.y >= *totalTiles) return;

    __shared__ float sA[MTILE * LDSS];    // 33280 B
    __shared__ int s_tok[MTILE];

    int4 td = tiles[blockIdx.y];
    int e = td.x, slot0 = td.y, hRow0 = td.z;
    int c = cnt[e];
    if (threadIdx.x < MTILE) {
        int slot = slot0 + (int)threadIdx.x;
        s_tok[threadIdx.x] = (slot < c) ? tokList[e * N_TOK + slot] : -1;
    }
    __syncthreads();

    // loader assignment: fixed row per thread, hoisted pointers
    int lr = threadIdx.x >> 3;               // 0..31
    int lc = (threadIdx.x & 7) << 2;         // 0,4,..,28
    int tok_r = s_tok[lr];
    const float* arow = x + (size_t)(tok_r < 0 ? 0 : tok_r) * EDIM;
    bool rowValid = tok_r >= 0;
    float* lrow = &sA[lr * LDSS + lc];

    int lane = threadIdx.x & 31, wave = threadIdx.x >> 5;
    int col  = blockIdx.x * 128 + wave * 16 + (lane & 15);
    bool colOK = col < HDIM;
    int colc = colOK ? col : (HDIM - 1);     // clamped, always in-bounds
    int koff = (lane < 16) ? 0 : 2;
    int ar   = lane & 15;
    const float* WvE = Wv + (size_t)e * EDIM * HDIM;
    const float* WgE = Wg + (size_t)e * EDIM * HDIM;

    v8f accV0 = {}, accV1 = {}, accG0 = {}, accG1 = {};

    for (int kb = 0; kb < EDIM; kb += KC) {
        if (kb) __syncthreads();             // LDS reuse guard
        if (rowValid) {
#pragma unroll
            for (int j = 0; j < 8; ++j) stage16(lrow + j * 32, arow + kb + lc + j * 32);
        } else {
            float4 z = make_float4(0.f, 0.f, 0.f, 0.f);
#pragma unroll
            for (int j = 0; j < 8; ++j) *(float4*)(lrow + j * 32) = z;
        }
        stage_wait();
        __syncthreads();
        for (int ks = 0; ks < KC; ks += 4) {
            const float* ap0 = &sA[ar * LDSS + ks + koff];
            const float* ap1 = ap0 + 16 * LDSS;
            v2f a0; a0.x = ap0[0]; a0.y = ap0[1];
            v2f a1; a1.x = ap1[0]; a1.y = ap1[1];
            int kk = kb + ks + koff;
            size_t base = (size_t)kk * HDIM + colc;
            v2f bV; bV.x = WvE[base]; bV.y = WvE[base + HDIM];
            v2f bG; bG.x = WgE[base]; bG.y = WgE[base + HDIM];
            accV0 = wmma_f32_4(a0, bV, accV0);
            accV1 = wmma_f32_4(a1, bV, accV1);
            accG0 = wmma_f32_4(a0, bG, accG0);
            accG1 = wmma_f32_4(a1, bG, accG1);
        }
    }

    float bvv = bv[e * HDIM + colc];
    float bgg = bg[e * HDIM + colc];
    int hi = (lane < 16) ? 0 : 8;
#pragma unroll
    for (int i = 0; i < 8; ++i) {
        float hv0 = accV0[i] + bvv, hg0 = accG0[i] + bgg;
        float hv1 = accV1[i] + bvv, hg1 = accG1[i] + bgg;
        float h0 = colOK ? (hv0 * (1.f / (1.f + expf(-hv0))) * hg0) : 0.f;
        float h1 = colOK ? (hv1 * (1.f / (1.f + expf(-hv1))) * hg1) : 0.f;
        hbuf[(size_t)(hRow0 + hi + i) * HPAD + col]      = h0;
        hbuf[(size_t)(hRow0 + 16 + hi + i) * HPAD + col] = h1;
    }
}

// ---------------------------------------------------------------------------
// Kernel 4: pass 2. oe = h @ Wo + bo over K=HPAD. A rows k>=H are exactly
// zero (zero-padded by pass 1). First 10 chunks need no Wo clamp; only the
// final chunk clamps row addresses (0 * finite = 0 keeps it exact).
// ---------------------------------------------------------------------------
__global__ __launch_bounds__(256) void moe_pass2(
    const float* __restrict__ hbuf, const float* __restrict__ Wo,
    const float* __restrict__ bo,
    const int* __restrict__ cnt, const int* __restrict__ tokList,
    const float* __restrict__ wList,
    const int4* __restrict__ tiles, const int* __restrict__ totalTiles,
    float* __restrict__ moe_out)
{
    if ((int)blockIdx.y >= *totalTiles) return;

    __shared__ float sA[MTILE * LDSS];
    __shared__ int   s_tok[MTILE];
    __shared__ float s_w[MTILE];

    int4 td = tiles[blockIdx.y];
    int e = td.x, slot0 = td.y, hRow0 = td.z;
    int c = cnt[e];
    if (threadIdx.x < MTILE) {
        int slot = slot0 + (int)threadIdx.x;
        bool ok = slot < c;
        s_tok[threadIdx.x] = ok ? tokList[e * N_TOK + slot] : -1;
        s_w[threadIdx.x]   = ok ? wList[e * N_TOK + slot] : 0.f;
    }

    // loader: fixed row per thread, pointer hoisted (hbuf rows always valid)
    int lr = threadIdx.x >> 3;
    int lc = (threadIdx.x & 7) << 2;
    const float* hrow = hbuf + (size_t)(hRow0 + lr) * HPAD + lc;
    float* lrow = &sA[lr * LDSS + lc];

    int lane = threadIdx.x & 31, wave = threadIdx.x >> 5;
    int col  = blockIdx.x * 128 + wave * 16 + (lane & 15);   // < 1024 always
    int koff = (lane < 16) ? 0 : 2;
    int ar   = lane & 15;
    const float* WoE = Wo + (size_t)e * HDIM * EDIM;

    v8f acc0 = {}, acc1 = {};

    for (int kb = 0; kb < HPAD; kb += KC) {      // 11 uniform chunks
        if (kb) __syncthreads();
#pragma unroll
        for (int j = 0; j < 8; ++j) stage16(lrow + j * 32, hrow + kb + j * 32);
        stage_wait();
        __syncthreads();
        if (kb + KC <= HDIM) {                   // fast path: no clamping
            for (int ks = 0; ks < KC; ks += 4) {
                const float* ap0 = &sA[ar * LDSS + ks + koff];
                const float* ap1 = ap0 + 16 * LDSS;
                v2f a0; a0.x = ap0[0]; a0.y = ap0[1];
                v2f a1; a1.x = ap1[0]; a1.y = ap1[1];
                size_t base = (size_t)(kb + ks + koff) * EDIM + col;
                v2f b; b.x = WoE[base]; b.y = WoE[base + EDIM];
                acc0 = wmma_f32_4(a0, b, acc0);
                acc1 = wmma_f32_4(a1, b, acc1);
            }
        } else {                                 // tail chunk: clamp rows
            for (int ks = 0; ks < KC; ks += 4) {
                const float* ap0 = &sA[ar * LDSS + ks + koff];
                const float* ap1 = ap0 + 16 * LDSS;
                v2f a0; a0.x = ap0[0]; a0.y = ap0[1];
                v2f a1; a1.x = ap1[0]; a1.y = ap1[1];
                int kk = kb + ks + koff;
                int k0 = (kk     < HDIM) ? kk     : (HDIM - 1);
                int k1 = (kk + 1 < HDIM) ? kk + 1 : (HDIM - 1);
                v2f b;
                b.x = WoE[(size_t)k0 * EDIM + col];
                b.y = WoE[(size_t)k1 * EDIM + col];
                acc0 = wmma_f32_4(a0, b, acc0);
                acc1 = wmma_f32_4(a1, b, acc1);
            }
        }
    }

    int hi = (lane < 16) ? 0 : 8;
    float bov = bo[e * EDIM + col];
#pragma unroll
    for (int i = 0; i < 8; ++i) {
        int r0 = hi + i, r1 = 16 + hi + i;
        int t0 = s_tok[r0]; float w0 = s_w[r0];
        int t1 = s_tok[r1]; float w1 = s_w[r1];
        if (t0 >= 0 && w0 != 0.f)
            atomicAdd(&moe_out[(size_t)t0 * EDIM + col], w0 * (acc0[i] + bov));
        if (t1 >= 0 && w1 != 0.f)
            atomicAdd(&moe_out[(size_t)t1 * EDIM + col], w1 * (acc1[i] + bov));
    }
}

// ---------------------------------------------------------------------------
// Kernel 5: residual + LayerNorm. One block (256 thr) per token, 4 elems/thr,
// fixed-order tree reduction (deterministic).
// ---------------------------------------------------------------------------
__global__ __launch_bounds__(256) void moe_ln(
    const float* __restrict__ x, const float* __restrict__ moe,
    const float* __restrict__ g, const float* __restrict__ b,
    float* __restrict__ out)
{
    int n = blockIdx.x;
    const float* xr = x   + (size_t)n * EDIM;
    const float* mr = moe + (size_t)n * EDIM;
    float y[4];
    float s = 0.f;
    __shared__ float red[256];
#pragma unroll
    for (int i = 0; i < 4; ++i) {
        int j = threadIdx.x + i * 256;
        float v = xr[j] + mr[j];
        y[i] = v; s += v;
    }
    red[threadIdx.x] = s;
    __syncthreads();
    for (int st = 128; st > 0; st >>= 1) {
        if (threadIdx.x < st) red[threadIdx.x] += red[threadIdx.x + st];
        __syncthreads();
    }
    float mu = red[0] * (1.f / EDIM);
    __syncthreads();
    float ss = 0.f;
#pragma unroll
    for (int i = 0; i < 4; ++i) { float d = y[i] - mu; ss += d * d; }
    red[threadIdx.x] = ss;
    __syncthreads();
    for (int st = 128; st > 0; st >>= 1) {
        if (threadIdx.x < st) red[threadIdx.x] += red[threadIdx.x + st];
        __syncthreads();
    }
    float inv = rsqrtf(red[0] * (1.f / EDIM) + 1e-5f);
#pragma unroll
    for (int i = 0; i < 4; ++i) {
        int j = threadIdx.x + i * 256;
        out[(size_t)n * EDIM + j] = g[j] * (y[i] - mu) * inv + b[j];
    }
}

// ---------------------------------------------------------------------------
extern "C" void kernel_launch(void* const* d_in, const int* in_sizes, int n_in,
                              void* d_out, int out_size, void* d_ws, size_t ws_size,
                              hipStream_t stream) {
    const float* x   = (const float*)d_in[0];
    const float* Wr  = (const float*)d_in[1];
    const float* Wv  = (const float*)d_in[2];
    const float* bv  = (const float*)d_in[3];
    const float* Wg  = (const float*)d_in[4];
    const float* bg  = (const float*)d_in[5];
    const float* Wo  = (const float*)d_in[6];
    const float* bo  = (const float*)d_in[7];
    const float* lng = (const float*)d_in[8];
    const float* lnb = (const float*)d_in[9];
    float* out = (float*)d_out;

    char* ws = (char*)d_ws;
    float* moe      = (float*)(ws + OFF_MOE);
    int*   cnt      = (int*)  (ws + OFF_CNT);
    int*   tokList  = (int*)  (ws + OFF_TOK);
    float* wList    = (float*)(ws + OFF_W);
    int4*  tiles    = (int4*) (ws + OFF_TILE);
    int*   totalT   = (int*)  (ws + OFF_NT);
    float* hbuf     = (float*)(ws + OFF_H);

    (void)hipMemsetAsync(moe, 0, (size_t)N_TOK * EDIM * sizeof(float), stream);
    (void)hipMemsetAsync(cnt, 0, NEXP * sizeof(int), stream);

    moe_router<<<N_TOK / 8, 256, 0, stream>>>(x, Wr, cnt, tokList, wList);
    moe_setup_tiles<<<1, 32, 0, stream>>>(cnt, tiles, totalT);
    moe_pass1<<<dim3(HPAD / 128, MAX_TILES), 256, 0, stream>>>(
        x, Wv, Wg, bv, bg, cnt, tokList, tiles, totalT, hbuf);
    moe_pass2<<<dim3(EDIM / 128, MAX_TILES), 256, 0, stream>>>(
        hbuf, Wo, bo, cnt, tokList, wList, tiles, totalT, moe);
    moe_ln<<<N_TOK, 256, 0, stream>>>(x, moe, lng, lnb, out);
}